// SparkFieldNet_19997367730512
// MI455X (gfx1250) — compile-verified
//
#include <hip/hip_runtime.h>
#include <hip/hip_bf16.h>

#define NDIM   8192
#define KSP    256
#define NSPLIT 4
#define KSLICE (NDIM / NSPLIT)   // 2048 K per wave

typedef __attribute__((ext_vector_type(2))) float f32x2;
typedef __attribute__((ext_vector_type(8))) float f32x8;

__device__ __forceinline__ unsigned h32(unsigned x) {
    x ^= x >> 16; x *= 0x7feb352dU;
    x ^= x >> 15; x *= 0x846ca68bU;
    x ^= x >> 16; return x;
}
__device__ __forceinline__ float u01(unsigned x) {
    return ((h32(x) >> 8) + 0.5f) * (1.0f / 16777216.0f);
}
__device__ __forceinline__ float clip2(float v) {
    return fminf(fmaxf(v, -2.0f), 2.0f);
}

// ---------------------------------------------------------------------------
// Kernel A: fused  s_part[ks] = (W @ (s*0.95)) partial over one K slice  AND
//                  W_out = clip(W * 0.999, -2, 2)  for that slice.
// Single pass over W (256 MB read + 256 MB NT store). Each wave: 16 rows x
// 2048 K via V_WMMA_F32_16X16X4_F32, K split 4-ways for occupancy.
// B is masked by a float multiply (no exec divergence in the hot loop).
// ---------------------------------------------------------------------------
__global__ void __launch_bounds__(256)
spark_field_update(const float* __restrict__ W,
                   const float* __restrict__ s_in,
                   float* __restrict__ W_out,
                   float* __restrict__ s_part) {
    const int ks   = blockIdx.y;          // K-split index 0..3
    const int kbeg = ks * KSLICE;

    __shared__ float s_sh[KSLICE];        // 8 KB: decayed state slice
    for (int i = threadIdx.x; i < KSLICE; i += blockDim.x)
        s_sh[i] = s_in[kbeg + i] * 0.95f;
    __syncthreads();

    const int lane = threadIdx.x & 31;
    const int wave = threadIdx.x >> 5;
    const int rowBase = blockIdx.x * 128 + wave * 16;   // 16 rows per wave
    const int m    = lane & 15;           // A-matrix: lanes 0-15 = M 0-15
    const int half = lane >> 4;           // 0: K pair {0,1}; 1: K pair {2,3}
    const float sel = (m == 0) ? 1.0f : 0.0f;  // B column N=0 mask, hoisted

    const size_t rowOff = (size_t)(rowBase + m) * NDIM + (size_t)kbeg
                        + (size_t)(half * 2);
    const float* wp = W     + rowOff;
    float*       wo = W_out + rowOff;

    f32x8 c = {};
    for (int k0 = 0; k0 < KSLICE; k0 += 16) {
        __builtin_prefetch(wp + k0 + 1024, 0, 0);   // global_prefetch_b8 ahead

        // A tiles: 4x (16x4 f32) slices of W, streamed non-temporally
        f32x2 a0 = __builtin_nontemporal_load((const f32x2*)(wp + k0));
        f32x2 a1 = __builtin_nontemporal_load((const f32x2*)(wp + k0 + 4));
        f32x2 a2 = __builtin_nontemporal_load((const f32x2*)(wp + k0 + 8));
        f32x2 a3 = __builtin_nontemporal_load((const f32x2*)(wp + k0 + 12));

        // B tiles: s in column N=0 only; all lanes load (LDS broadcast),
        // non-column-0 lanes zeroed by multiply -> no exec-mask branches.
        f32x2 b0 = (*(const f32x2*)&s_sh[k0      + 2 * half]) * sel;
        f32x2 b1 = (*(const f32x2*)&s_sh[k0 +  4 + 2 * half]) * sel;
        f32x2 b2 = (*(const f32x2*)&s_sh[k0 +  8 + 2 * half]) * sel;
        f32x2 b3 = (*(const f32x2*)&s_sh[k0 + 12 + 2 * half]) * sel;

        // fused global decay + clip write-out (the "free" second W pass)
        f32x2 d0, d1, d2, d3;
        d0.x = clip2(a0.x * 0.999f); d0.y = clip2(a0.y * 0.999f);
        d1.x = clip2(a1.x * 0.999f); d1.y = clip2(a1.y * 0.999f);
        d2.x = clip2(a2.x * 0.999f); d2.y = clip2(a2.y * 0.999f);
        d3.x = clip2(a3.x * 0.999f); d3.y = clip2(a3.y * 0.999f);
        __builtin_nontemporal_store(d0, (f32x2*)(wo + k0));
        __builtin_nontemporal_store(d1, (f32x2*)(wo + k0 + 4));
        __builtin_nontemporal_store(d2, (f32x2*)(wo + k0 + 8));
        __builtin_nontemporal_store(d3, (f32x2*)(wo + k0 + 12));

        c = __builtin_amdgcn_wmma_f32_16x16x4_f32(false, a0, false, b0,
                                                  (short)0, c, false, false);
        c = __builtin_amdgcn_wmma_f32_16x16x4_f32(false, a1, false, b1,
                                                  (short)0, c, false, false);
        c = __builtin_amdgcn_wmma_f32_16x16x4_f32(false, a2, false, b2,
                                                  (short)0, c, false, false);
        c = __builtin_amdgcn_wmma_f32_16x16x4_f32(false, a3, false, b3,
                                                  (short)0, c, false, false);
    }

    // D layout (32-bit C/D 16x16): column N=0 lives in lanes 0 (M=0..7) and
    // 16 (M=8..15). Write this K-slice's partial sums (fixed-order reduce
    // later -> deterministic, no float atomics).
    if (m == 0) {
        for (int r = 0; r < 8; ++r) {
            int row = rowBase + half * 8 + r;
            s_part[(size_t)ks * NDIM + row] = c[r];
        }
    }
}

// ---------------------------------------------------------------------------
// Kernel B: sequential spark scan. One workgroup (1024 threads = 32 waves).
// Prologue reduces the 4 K-slice partials in fixed order, applies noise +
// sigmoid + spark pinning. s and M live in LDS (320 KB/WGP on CDNA5).
// Categorical sampling = Gumbel-max argmax (wave32 shuffle + LDS reduce).
// W edits kept in an LDS overlay, patched into W_out at the end.
// ---------------------------------------------------------------------------
__global__ void spark_scan(const float* __restrict__ W,
                           const float* __restrict__ s_part,
                           const float* __restrict__ M_in,
                           const int*   __restrict__ pos_in,
                           const float* __restrict__ en_in,
                           const int*   __restrict__ age_in,
                           float* __restrict__ W_out,
                           int*   __restrict__ pos_out,
                           float* __restrict__ s_out,
                           float* __restrict__ M_out,
                           float* __restrict__ en_out,
                           int*   __restrict__ age_out) {
    extern __shared__ float lds[];
    float* s_sh = lds;                 // NDIM
    float* M_sh = s_sh + NDIM;         // NDIM
    float* redV = M_sh + NDIM;         // 32
    int*   redI = (int*)(redV + 32);   // 32
    int*   ovR  = redI + 32;           // KSP  (overlay rows)
    int*   ovC  = ovR + KSP;           // KSP  (overlay cols)
    float* ovV  = (float*)(ovC + KSP); // KSP  (overlay values)

    const int tid  = threadIdx.x;
    const int nthr = blockDim.x;
    const int lane = tid & 31;
    const int wv   = tid >> 5;
    const int nwaves = nthr >> 5;

    for (int i = tid; i < NDIM; i += nthr) {
        float acc = s_part[i] + s_part[NDIM + i]
                  + s_part[2 * NDIM + i] + s_part[3 * NDIM + i];
        unsigned hx = (unsigned)i * 2u + 0x9e3779b9u;
        float u1 = u01(hx);
        float u2 = u01(hx ^ 0x85ebca6bu);
        float z  = sqrtf(-2.0f * __logf(u1)) * __cosf(6.28318530718f * u2);
        float v  = acc + 0.05f * z;                // NOISE_STD
        s_sh[i] = 1.0f / (1.0f + __expf(-v));      // sigmoid
        M_sh[i] = M_in[i] * 0.95f;                 // M_DECAY
    }
    __syncthreads();
    // force: sparks younger than SPARK_FORCE_STEPS pin their node to 1.0
    for (int i = tid; i < KSP; i += nthr)
        if (age_in[i] < 5) s_sh[pos_in[i]] = 1.0f;
    __syncthreads();

    for (int i = 0; i < KSP; ++i) {
        const int prev = pos_in[i];

        // Gumbel-max categorical over relu(W[prev])/TEMP + 1e-6/TEMP + 0.8*M
        float best = -1e30f; int bestIdx = 0;
        for (int col = tid; col < NDIM; col += nthr) {
            float w = W[(size_t)prev * NDIM + col];
            for (int j = 0; j < i; ++j)            // overlay: prior scan edits
                if (ovR[j] == prev && ovC[j] == col) w = ovV[j];
            float logit = (fmaxf(w, 0.0f) + 1e-6f) * (1.0f / 0.3f)
                        + 0.8f * M_sh[col];
            unsigned hx = ((unsigned)i * 9176u + (unsigned)col) * 2654435761u
                        + 0xB5297A4Du;
            float g = -__logf(-__logf(u01(hx)));
            float sc = logit + g;
            if (sc > best) { best = sc; bestIdx = col; }
        }
        for (int off = 16; off > 0; off >>= 1) {   // wave32 argmax
            float ov = __shfl_down(best, off, 32);
            int   oi = __shfl_down(bestIdx, off, 32);
            if (ov > best) { best = ov; bestIdx = oi; }
        }
        if (lane == 0) { redV[wv] = best; redI[wv] = bestIdx; }
        __syncthreads();

        if (tid == 0) {
            float b2 = redV[0]; int bi = redI[0];
            for (int w2 = 1; w2 < nwaves; ++w2)
                if (redV[w2] > b2) { b2 = redV[w2]; bi = redI[w2]; }

            int nxt = bi;
            float ue = u01(0xC2B2AE35u ^ ((unsigned)i * 0x27d4eb2fu));
            if (ue < 0.05f) {                      // EXPLORE_CHANCE
                unsigned hr = h32(0x165667B1u + (unsigned)i * 0x9E3779B1u);
                nxt = (int)(hr & (NDIM - 1));
            }

            // edge update W[nxt,prev] = old*(1-LR_EDGE) + s[prev]*LR_EDGE
            float wold = W[(size_t)nxt * NDIM + prev];
            for (int j = 0; j < i; ++j)
                if (ovR[j] == nxt && ovC[j] == prev) wold = ovV[j];
            float wnew = wold * 0.95f + s_sh[prev] * 0.05f;
            ovR[i] = nxt; ovC[i] = prev; ovV[i] = wnew;

            M_sh[nxt] += 0.2f;                     // M_DEPOSIT
            float en = en_in[i] * 0.98f;           // SPARK_ENERGY_DECAY
            s_sh[nxt] = en;
            int ag = age_in[i] + 1;
            int p  = nxt;
            if (en < 0.05f) { p = i; en = 1.0f; ag = 0; }  // reset (i % n == i)
            pos_out[i] = p; en_out[i] = en; age_out[i] = ag;
        }
        __syncthreads();
    }

    for (int i = tid; i < NDIM; i += nthr) {
        s_out[i] = s_sh[i];
        M_out[i] = M_sh[i];
    }
    if (tid == 0) {                                // patch decayed W_out
        for (int j = 0; j < KSP; ++j)
            W_out[(size_t)ovR[j] * NDIM + ovC[j]] = clip2(ovV[j] * 0.999f);
    }
}

extern "C" void kernel_launch(void* const* d_in, const int* in_sizes, int n_in,
                              void* d_out, int out_size, void* d_ws, size_t ws_size,
                              hipStream_t stream) {
    const float* W   = (const float*)d_in[0];
    const float* s   = (const float*)d_in[1];
    const float* M   = (const float*)d_in[2];
    const int*   pos = (const int*)  d_in[3];
    const float* en  = (const float*)d_in[4];
    const int*   age = (const int*)  d_in[5];

    // d_out layout (return order): pos[K] i32, W[N*N] f32, s[N] f32,
    //                              M[N] f32, energy[K] f32, age[K] i32
    float* out = (float*)d_out;
    const size_t NN = (size_t)NDIM * NDIM;
    int*   pos_out = (int*)out;
    float* W_out   = out + KSP;
    float* s_out   = W_out + NN;
    float* M_out   = s_out + NDIM;
    float* en_out  = M_out + NDIM;
    int*   age_out = (int*)(en_out + KSP);

    float* s_part = (float*)d_ws;   // NSPLIT * NDIM floats of K-slice partials

    // Pass 1: fused matvec (WMMA f32 16x16x4) + W decay/clip NT write-out.
    // 64 row-blocks x 4 K-splits, 8 waves each -> 2048 waves in flight.
    spark_field_update<<<dim3(NDIM / 128, NSPLIT), 256, 0, stream>>>(
        W, s, W_out, s_part);

    // Pass 2: sequential spark scan, single WGP, s/M resident in LDS.
    size_t shmem = (size_t)(2 * NDIM + 32) * sizeof(float)
                 + 32 * sizeof(int)
                 + (size_t)KSP * (2 * sizeof(int) + sizeof(float));
    spark_scan<<<1, 1024, shmem, stream>>>(W, s_part, M, pos, en, age,
                                           W_out, pos_out, s_out, M_out,
                                           en_out, age_out);
}